// RandomSpatial_86311662780443
// MI455X (gfx1250) — compile-verified
//
#include <hip/hip_runtime.h>

typedef __attribute__((ext_vector_type(2))) float v2f;
typedef __attribute__((ext_vector_type(4))) float v4f;
typedef __attribute__((ext_vector_type(8))) float v8f;

#define B_DIM 16
#define H_DIM 512
#define W_DIM 512
#define C_DIM 16
#define NPIX  (B_DIM * H_DIM * W_DIM)   // 4,194,304 (exact multiple of 256)

// One thread = one output pixel (all 16 channels, processed as 4x float4).
// Coordinate transform gx,gy = theta * [x;y;1] done with V_WMMA_F32_16X16X4_F32:
//   A (16x4): row0 = (t00,t01,t02,0), row1 = (t10,t11,t12,0), rows 2..15 = 0
//   B (4x16): K0 = x_t(pixel n), K1 = y_t(pixel n), K2 = 1, K3 = 0
//   D (16x16): D[0,n] = gx(n) lands in lane n's d[0]; D[1,n] = gy(n) in d[1].
// Two WMMAs per wave cover 32 pixels; shuffle hands group-1 results to lanes 16..31.
__global__ __launch_bounds__(256) void RandomSpatial_bilinear_wmma(
    const float* __restrict__ img,    // (B,H,W,C)
    const float* __restrict__ theta,  // (B,2,3)
    float* __restrict__ out)          // (B,H,W,C)
{
    const int n    = blockIdx.x * 256 + (int)threadIdx.x;
    const int lane = (int)threadIdx.x & 31;

    const int b   = n >> 18;          // H*W = 2^18
    const int rem = n & 0x3FFFF;
    const int py  = rem >> 9;         // W = 2^9
    const int px  = rem & 511;

    // theta is uniform across the block -> scalar loads
    const int bs = __builtin_amdgcn_readfirstlane(b);
    const float* th = theta + bs * 6;
    const float t00 = th[0], t01 = th[1], t02 = th[2];
    const float t10 = th[3], t11 = th[4], t12 = th[5];

    // linspace(-1,1,512): step = 2/511
    const float xt = fmaf((float)px, 2.0f / 511.0f, -1.0f);
    const float yt = fmaf((float)py, 2.0f / 511.0f, -1.0f);

    // ---- A matrix (16x4 f32 -> 2 VGPRs). ISA layout:
    // v0: lanes 0-15 = A[m=lane,K=0], lanes 16-31 = A[m=lane-16,K=2]
    // v1: lanes 0-15 = A[m=lane,K=1], lanes 16-31 = A[m=lane-16,K=3]
    float a0 = 0.0f, a1 = 0.0f;
    if      (lane == 0)  { a0 = t00; a1 = t01; }
    else if (lane == 1)  { a0 = t10; a1 = t11; }
    else if (lane == 16) { a0 = t02; }
    else if (lane == 17) { a0 = t12; }
    v2f A; A.x = a0; A.y = a1;

    // ---- B matrices (4x16 f32 -> 2 VGPRs). ISA layout:
    // v0: lanes 0-15 = B[K=0,n=lane] (x), lanes 16-31 = B[K=2,n] (ones)
    // v1: lanes 0-15 = B[K=1,n=lane] (y), lanes 16-31 = B[K=3,n] (zero)
    const bool lo = (lane < 16);
    v2f B0; B0.x = lo ? xt : 1.0f;  B0.y = lo ? yt : 0.0f;

    const float xt_hi = __shfl(xt, lane | 16);   // lanes 0-15 fetch upper partner's coords
    const float yt_hi = __shfl(yt, lane | 16);
    v2f B1; B1.x = lo ? xt_hi : 1.0f;  B1.y = lo ? yt_hi : 0.0f;

    v8f cz = {};
    v8f d0 = __builtin_amdgcn_wmma_f32_16x16x4_f32(false, A, false, B0,
                                                   (short)0, cz, false, false);
    v8f d1 = __builtin_amdgcn_wmma_f32_16x16x4_f32(false, A, false, B1,
                                                   (short)0, cz, false, false);

    // lanes 0-15: own pixel's gx/gy already in d0[0]/d0[1].
    // lanes 16-31: pixel (lane) == group-1 pixel (lane-16) -> shuffle from lane&15.
    const float gxh = __shfl(d1[0], lane & 15);
    const float gyh = __shfl(d1[1], lane & 15);
    const float gx = lo ? d0[0] : gxh;
    const float gy = lo ? d0[1] : gyh;

    // x = 0.5*((gx+1)*(W-2)) = (gx+1)*255 ; same for y
    const float x = (gx + 1.0f) * 255.0f;
    const float y = (gy + 1.0f) * 255.0f;

    int x0 = (int)floorf(x);
    int y0 = (int)floorf(y);
    x0 = min(max(x0, 0), W_DIM - 1);
    y0 = min(max(y0, 0), H_DIM - 1);
    const int x1 = min(x0 + 1, W_DIM - 1);
    const int y1 = min(y0 + 1, H_DIM - 1);

    const float x0f = (float)x0, x1f = (float)x1;
    const float y0f = (float)y0, y1f = (float)y1;
    const float wx1 = x1f - x, wx0 = x - x0f;
    const float wy1 = y1f - y, wy0 = y - y0f;
    const float wa = wx1 * wy1;   // (y0,x0)
    const float wb = wx1 * wy0;   // (y1,x0)
    const float wc = wx0 * wy1;   // (y0,x1)
    const float wd = wx0 * wy0;   // (y1,x1)

    const size_t rowB = (size_t)b * H_DIM;
    const float* pa = img + (((rowB + y0) * W_DIM) + x0) * C_DIM;
    const float* pb = img + (((rowB + y1) * W_DIM) + x0) * C_DIM;
    const float* pc = img + (((rowB + y0) * W_DIM) + x1) * C_DIM;
    const float* pd = img + (((rowB + y1) * W_DIM) + x1) * C_DIM;
    float* po = out + (size_t)n * C_DIM;

#pragma unroll
    for (int q = 0; q < 4; ++q) {
        // temporal 128-bit loads: exploit the 192MB L2 for texel reuse
        const v4f Ia = *(const v4f*)(pa + 4 * q);
        const v4f Ib = *(const v4f*)(pb + 4 * q);
        const v4f Ic = *(const v4f*)(pc + 4 * q);
        const v4f Id = *(const v4f*)(pd + 4 * q);
        v4f r;
        r.x = fmaf(wa, Ia.x, fmaf(wb, Ib.x, fmaf(wc, Ic.x, wd * Id.x)));
        r.y = fmaf(wa, Ia.y, fmaf(wb, Ib.y, fmaf(wc, Ic.y, wd * Id.y)));
        r.z = fmaf(wa, Ia.z, fmaf(wb, Ib.z, fmaf(wc, Ic.z, wd * Id.z)));
        r.w = fmaf(wa, Ia.w, fmaf(wb, Ib.w, fmaf(wc, Ic.w, wd * Id.w)));
        // write-once 256MB output stream: non-temporal so it doesn't evict img from L2
        __builtin_nontemporal_store(r, (v4f*)(po + 4 * q));
    }
}

extern "C" void kernel_launch(void* const* d_in, const int* in_sizes, int n_in,
                              void* d_out, int out_size, void* d_ws, size_t ws_size,
                              hipStream_t stream) {
    (void)in_sizes; (void)n_in; (void)out_size; (void)d_ws; (void)ws_size;
    const float* img   = (const float*)d_in[0];
    const float* theta = (const float*)d_in[1];
    float*       out   = (float*)d_out;

    const int threads = 256;                 // 8 wave32 waves per block
    const int blocks  = NPIX / threads;      // 16384, exact -> EXEC all-1s at WMMA
    RandomSpatial_bilinear_wmma<<<blocks, threads, 0, stream>>>(img, theta, out);
}